// ProteinPooler_39754217292264
// MI455X (gfx1250) — compile-verified
//
#include <hip/hip_runtime.h>
#include <hip/hip_bf16.h>

// ProteinPooler: segment-mean pooling, [N,L,D] f32 + [N] keys -> [B,D] f32.
// Strategy: per-protein column sums on the WMMA f32 pipe (A = ones trick),
// global f32 atomics into a ws accumulator, then divide by token counts.

typedef float v2f __attribute__((ext_vector_type(2)));
typedef float v8f __attribute__((ext_vector_type(8)));

#define L_SEQ 512
#define D_DIM 256
#define B_SEG 256

// ---------------------------------------------------------------------------
// Kernel 1: one workgroup per protein. 8 waves; wave w reduces columns
// [32w, 32w+32) over all L rows using V_WMMA_F32_16X16X4_F32 with A = ones:
//   D = ones(16x4) x B(4x16) + C  ->  every row of D = colsum of the B tile.
// Exact f32 accumulation, so result matches the reference's f32 segment_sum.
// ---------------------------------------------------------------------------
__global__ __launch_bounds__(256) void pool_sum_kernel(
    const float* __restrict__ X,     // [N, L, D]
    const int*   __restrict__ keys,  // [N]
    float*       __restrict__ acc)   // [B, D] f32 accumulator (zeroed)
{
    const int p    = blockIdx.x;        // protein index
    const int tid  = threadIdx.x;       // 0..255
    const int wave = tid >> 5;          // 0..7
    const int lane = tid & 31;          // wave32
    const int dbase = wave * 32;        // this wave's 32 feature columns
    const int n    = lane & 15;         // column within a 16-wide tile
    const int klo  = (lane >> 4) << 1;  // lanes 0-15 -> K rows {0,1}; 16-31 -> {2,3}

    const float* base = X + (size_t)p * (L_SEQ * D_DIM);

    // A = ones(16x4): 2 VGPRs, all lanes = 1.0f. With A==1 the K<->lane
    // striping of B is irrelevant to the sum; only coverage matters.
    v2f a; a[0] = 1.0f; a[1] = 1.0f;

    v8f c0 = {};  // accumulates colsums for columns [dbase,      dbase+16)
    v8f c1 = {};  // accumulates colsums for columns [dbase+16,   dbase+32)

    // Lane's element addresses: column fixed, rows advance by 4 each step.
    const float* p0 = base + (size_t)klo * D_DIM + dbase + n;   // tile 0
    const float* p1 = p0 + 16;                                  // tile 1

    #pragma unroll 4
    for (int k = 0; k < L_SEQ; k += 4) {
        // Each load instruction: two contiguous 64B segments (fully coalesced).
        // Non-temporal: 1 GiB streamed once, keep it out of L2 residency.
        v2f b0, b1;
        b0[0] = __builtin_nontemporal_load(p0);
        b0[1] = __builtin_nontemporal_load(p0 + D_DIM);
        b1[0] = __builtin_nontemporal_load(p1);
        b1[1] = __builtin_nontemporal_load(p1 + D_DIM);
        p0 += 4 * D_DIM;
        p1 += 4 * D_DIM;

        // 8-arg form: (neg_a, A, neg_b, B, c_mod, C, reuse_a, reuse_b)
        c0 = __builtin_amdgcn_wmma_f32_16x16x4_f32(
                 false, a, false, b0, (short)0, c0, false, false);
        c1 = __builtin_amdgcn_wmma_f32_16x16x4_f32(
                 false, a, false, b1, (short)0, c1, false, false);
    }

    // C/D layout (16x16 f32): VGPR0, lanes 0-15 = row M=0 cols N=0-15;
    // lanes 16-31 = row M=8 cols N=0-15. All rows identical (A==ones), so:
    //   lanes  0-15: c0[0] = colsum[dbase + lane]
    //   lanes 16-31: c1[0] = colsum[dbase + 16 + (lane-16)] = colsum[dbase + lane]
    float s  = (lane < 16) ? c0[0] : c1[0];
    int  col = dbase + lane;

    const int b = keys[p];
    unsafeAtomicAdd(&acc[(size_t)b * D_DIM + col], s);  // global_atomic_add_f32
}

// ---------------------------------------------------------------------------
// Kernel 2: block b computes count(keys==b), writes mean = sum/(count*L).
// ---------------------------------------------------------------------------
__global__ __launch_bounds__(256) void finalize_kernel(
    const int*   __restrict__ keys,  // [N]
    const float* __restrict__ acc,   // [B, D] sums
    float*       __restrict__ out,   // [B, D]
    int N)
{
    __shared__ int s_count;
    const int b = blockIdx.x;
    const int d = threadIdx.x;

    if (d == 0) s_count = 0;
    __syncthreads();

    int cnt = 0;
    for (int i = d; i < N; i += 256) cnt += (keys[i] == b) ? 1 : 0;
    atomicAdd(&s_count, cnt);
    __syncthreads();

    const float denom = (float)s_count * (float)L_SEQ;
    out[(size_t)b * D_DIM + d] = acc[(size_t)b * D_DIM + d] / denom;
}

extern "C" void kernel_launch(void* const* d_in, const int* in_sizes, int n_in,
                              void* d_out, int out_size, void* d_ws, size_t ws_size,
                              hipStream_t stream) {
    const float* X    = (const float*)d_in[0];  // [N, L, D] f32
    const int*   keys = (const int*)d_in[1];    // [N] int
    float*       out  = (float*)d_out;          // [B, D] f32
    float*       acc  = (float*)d_ws;           // [B, D] f32 scratch

    const int N = in_sizes[1];                  // 2048 proteins

    // Zero the accumulator every call (graph replays must be deterministic).
    hipMemsetAsync(acc, 0, (size_t)B_SEG * D_DIM * sizeof(float), stream);

    pool_sum_kernel<<<N, 256, 0, stream>>>(X, keys, acc);
    finalize_kernel<<<B_SEG, 256, 0, stream>>>(keys, acc, out, N);
}